// ParametrisedCompatibility_29472065585921
// MI455X (gfx1250) — compile-verified
//
#include <hip/hip_runtime.h>

typedef __attribute__((ext_vector_type(2))) float v2f;
typedef __attribute__((ext_vector_type(8))) float v8f;

#define C_DIM 512
#define WAVES_PER_BLOCK 8
#define PIX_PER_WAVE 16
#define PIX_PER_BLOCK (WAVES_PER_BLOCK * PIX_PER_WAVE)

// out[p] = sum_c (l[p,c] + g[p,c]) * u[c]   for p in [0, npix)
// One wave32 handles 16 pixels via V_WMMA_F32_16X16X4_F32:
//   A (16x4 f32)  = (l+g) tile: lanes 0-15 -> row M=lane, K={k,k+1};
//                              lanes 16-31 -> row M=lane-16, K={k+2,k+3}
//   B (4x16 f32)  = u[k..k+3] broadcast across all 16 columns (mirrored layout)
//   D (16x16 f32) = every column holds the 16 per-pixel dot products
// l/g are streamed once (411 MB >> 192 MB L2) -> non-temporal loads; out is
// write-once -> non-temporal store. u lives in LDS.
__global__ __launch_bounds__(256) void pc_dot_wmma_kernel(
    const float* __restrict__ l, const float* __restrict__ g,
    const float* __restrict__ u, float* __restrict__ out, int npix)
{
    __shared__ float uS[C_DIM];
    for (int i = threadIdx.x; i < C_DIM; i += blockDim.x) uS[i] = u[i];
    __syncthreads();

    const int lane = threadIdx.x & 31;            // wave32
    const int wave = threadIdx.x >> 5;
    const int p0   = (blockIdx.x * WAVES_PER_BLOCK + wave) * PIX_PER_WAVE;
    if (p0 >= npix) return;                       // whole-wave uniform exit

    const int m    = lane & 15;                   // pixel row within tile
    const int koff = (lane >> 4) << 1;            // 0 (lanes 0-15) or 2 (lanes 16-31)

    const float* lrow = l + (size_t)(p0 + m) * C_DIM + koff;
    const float* grow = g + (size_t)(p0 + m) * C_DIM + koff;
    const float* ubuf = uS + koff;

    v8f acc = {};
    #pragma unroll 8
    for (int k = 0; k < C_DIM; k += 4) {
        const v2f lv = __builtin_nontemporal_load((const v2f*)(lrow + k)); // global_load_b64 th:NT
        const v2f gv = __builtin_nontemporal_load((const v2f*)(grow + k)); // global_load_b64 th:NT
        const v2f a  = lv + gv;                   // v_pk_add_f32
        const v2f b  = *(const v2f*)(ubuf + k);   // ds_load_b64 (broadcast)
        // 8 args: (neg_a, A, neg_b, B, c_mod, C, reuse_a, reuse_b)
        acc = __builtin_amdgcn_wmma_f32_16x16x4_f32(
            /*neg_a=*/false, a, /*neg_b=*/false, b,
            /*c_mod=*/(short)0, acc, /*reuse_a=*/false, /*reuse_b=*/false);
    }

    // D: VGPR v holds M=v (lanes 0-15) / M=v+8 (lanes 16-31); all columns equal.
    if ((lane & 15) == 0) {
        float* dst = out + p0 + ((lane >> 4) << 3);  // lane0 -> +0, lane16 -> +8
        #pragma unroll
        for (int i = 0; i < 8; ++i)
            __builtin_nontemporal_store(acc[i], dst + i);
    }
}

extern "C" void kernel_launch(void* const* d_in, const int* in_sizes, int n_in,
                              void* d_out, int out_size, void* d_ws, size_t ws_size,
                              hipStream_t stream) {
    const float* l = (const float*)d_in[0];   // (B,H,W,C) f32
    const float* g = (const float*)d_in[1];   // (B,H,W,C) f32
    const float* u = (const float*)d_in[2];   // (C,1)     f32
    float* out = (float*)d_out;               // (B,H,W)   f32

    const int npix = out_size;                // B*H*W = 100352
    const int blocks = (npix + PIX_PER_BLOCK - 1) / PIX_PER_BLOCK;  // 784

    hipLaunchKernelGGL(pc_dot_wmma_kernel, dim3(blocks), dim3(256), 0, stream,
                       l, g, u, out, npix);
}